// MLSWATNeuron_80058190397905
// MI455X (gfx1250) — compile-verified
//
#include <hip/hip_runtime.h>
#include <math.h>

// MLSWAT spiking neuron, T=4 timesteps, L=4 thresholds.
// Memory-bound streaming kernel: ~537 MB traffic -> ~23us floor @ 23.3 TB/s.
// CDNA5 path: global_load_async_to_lds_b128 (th:TH_LOAD_NT) double-buffered
// prefetch with ASYNCcnt pipelining + non-temporal b128 stores, so neither the
// 256 MiB input stream nor the 256 MiB output stream churns the 192 MB L2.

typedef __attribute__((ext_vector_type(4))) float f4;

#define TSTEPS 4
#define LTH 4
#define TPB 256                   // 8 waves (wave32) per block
#define TILE_FLOATS (TPB * 4)     // 1024 floats per plane per tile
#define MAX_BLOCKS 4096

// ---------------- prologue: compute the 6 scalar constants ----------------
__global__ void mlswat_consts_kernel(const float* __restrict__ raw,
                                     const float* __restrict__ zp,
                                     float* __restrict__ cons) {
    float inc[LTH], cum[LTH], th[LTH];
#pragma unroll
    for (int i = 0; i < LTH; ++i) {
        float r = raw[i];
        // jax.nn.softplus(x) = max(x,0) + log1p(exp(-|x|))
        inc[i] = fmaxf(r, 0.0f) + log1pf(expf(-fabsf(r)));
    }
    cum[0] = inc[0];
#pragma unroll
    for (int i = 1; i < LTH; ++i) cum[i] = cum[i - 1] + inc[i];
#pragma unroll
    for (int i = 0; i < LTH; ++i) {
        float v = cum[LTH - 1 - i];                 // flip -> descending
        th[i] = fminf(fmaxf(v, 1e-4f), 1e4f);       // clamp (STE fwd)
    }
    // rzp = clip(round_half_even(zp), 0, 15); tiled to (L,)
    float rzp = fminf(fmaxf(rintf(zp[0]), 0.0f), 15.0f);
    float term = (rzp / (float)TSTEPS) * ((th[0] - th[LTH - 1]) / (float)LTH);
    float Iz = 0.0f;
#pragma unroll
    for (int i = 0; i < LTH; ++i) Iz += term;       // .sum() over L equal terms

    cons[0] = th[0]; cons[1] = th[1]; cons[2] = th[2]; cons[3] = th[3];
    cons[4] = Iz;
    cons[5] = th[0] * 0.5f;                         // v0
}

// ---------------- main streaming kernel ----------------
__global__ void __launch_bounds__(TPB)
mlswat_stream_kernel(const float* __restrict__ x, float* __restrict__ out,
                     const float* __restrict__ cons,
                     unsigned planeElems, unsigned numTiles) {
    // 2 buffers x 4 timestep planes x 256 lanes x 16B = 32 KB LDS
    __shared__ f4 smem[2][TSTEPS][TPB];

    const unsigned tid = threadIdx.x;
    const float th0 = cons[0], th1 = cons[1], th2 = cons[2], th3 = cons[3];
    const float Iz = cons[4], v0 = cons[5];
    const unsigned planeBytes = planeElems * 4u;           // 64 MiB
    const unsigned ldsBase = (unsigned)(size_t)(&smem[0][0][0]);
    const unsigned laneSlot = tid * 16u;

    // Issue 4 async B128 loads (one per timestep plane) for a tile into buf.
    // GVS mode: saddr = x base (SGPR pair), vaddr = 32-bit byte offset.
    // th:TH_LOAD_NT -> streaming data, do not retain in L2 (no reuse).
    auto issue = [&](unsigned tile, int buf) {
        unsigned gOff = tile * (unsigned)(TILE_FLOATS * 4) + laneSlot;
#pragma unroll
        for (int p = 0; p < TSTEPS; ++p) {
            unsigned lds = ldsBase + ((unsigned)(buf * TSTEPS + p) * TPB) * 16u + laneSlot;
            unsigned ga = gOff + (unsigned)p * planeBytes;
            asm volatile("global_load_async_to_lds_b128 %0, %1, %2 th:TH_LOAD_NT"
                         :: "v"(lds), "v"(ga), "s"(x)
                         : "memory");
        }
    };

    unsigned tile = blockIdx.x;
    int buf = 0;
    issue(tile, 0);                                        // prime the pipeline

    for (; tile < numTiles; tile += gridDim.x) {
        unsigned nxt = tile + gridDim.x;
        if (nxt < numTiles) {
            issue(nxt, buf ^ 1);                           // prefetch next tile
            // 8 async ops outstanding; wait until current tile's 4 landed
            // (async loads complete in order), leaving the prefetch in flight.
            asm volatile("s_wait_asynccnt 0x4" ::: "memory");
        } else {
            asm volatile("s_wait_asynccnt 0x0" ::: "memory");
        }

        const unsigned elem = tile * (unsigned)TILE_FLOATS + tid * 4u;
        float v[4] = {v0, v0, v0, v0};
#pragma unroll
        for (int t = 0; t < TSTEPS; ++t) {
            f4 xv = smem[buf][t][tid];                     // ds_load_b128 (own slot)
            f4 ov;
#pragma unroll
            for (int c = 0; c < 4; ++c) {
                float vv = v[c] + (xv[c] + Iz);            // v = sg(v) + sg(x+Iz)
                // exclusive multi-threshold spike: largest threshold exceeded
                float spk = (vv >= th0) ? th0
                          : (vv >= th1) ? th1
                          : (vv >= th2) ? th2
                          : (vv >= th3) ? th3 : 0.0f;
                v[c] = vv - spk;
                ov[c] = spk - Iz;
            }
            // streaming store, bypass L2 retention (no reuse of output)
            __builtin_nontemporal_store(ov, (f4*)(out + (size_t)t * planeElems + elem));
        }
        buf ^= 1;
    }
}

extern "C" void kernel_launch(void* const* d_in, const int* in_sizes, int n_in,
                              void* d_out, int out_size, void* d_ws, size_t ws_size,
                              hipStream_t stream) {
    const float* x   = (const float*)d_in[0];   // (T, 4096, 4096) f32
    const float* raw = (const float*)d_in[1];   // (L,) f32
    const float* zp  = (const float*)d_in[2];   // (1,) f32
    float* out  = (float*)d_out;                // (T, 4096, 4096) f32
    float* cons = (float*)d_ws;                 // 6 scalar constants

    const unsigned total = (unsigned)in_sizes[0];        // T * 4096 * 4096
    const unsigned plane = total / TSTEPS;               // 16,777,216
    const unsigned numTiles = plane / TILE_FLOATS;       // 16,384
    const unsigned blocks = numTiles < MAX_BLOCKS ? numTiles : MAX_BLOCKS;

    hipLaunchKernelGGL(mlswat_consts_kernel, dim3(1), dim3(1), 0, stream,
                       raw, zp, cons);
    hipLaunchKernelGGL(mlswat_stream_kernel, dim3(blocks), dim3(TPB), 0, stream,
                       x, out, cons, plane, numTiles);
}